// SimpleTransformerWrapper_29472065585905
// MI455X (gfx1250) — compile-verified
//
#include <hip/hip_runtime.h>

typedef __attribute__((ext_vector_type(2))) float v2f;
typedef __attribute__((ext_vector_type(8))) float v8f;

#define B_TOT 4096
#define SEQ   128
#define DIN   64
#define DOUT  128
#define HID   32
#define NOUT  4
#define BPW   8   // batches per wave (2 gathered rows each -> 16 M-rows)

// One wave32 per block. Per wave:
//   Phase 1: lanes 0..7 scan one mask row each (u64 popcount/ffs) -> i0, i1, n_valid
//   Phase 2: gathered A(16x64) @ W_net(64x128) via 128x v_wmma_f32_16x16x4_f32
//   Phase 3: +b_net, C fragments -> LDS feats[16][128]
//   Phase 4: VALU MLP: h = relu(feats@W1+b1), preds = h@W2+b2
//   Phase 5: n_valid gating, store (B,2,4)
__global__ __launch_bounds__(32)
void fused_topk_transformer_kernel(const float* __restrict__ x,
                                   const float* __restrict__ Wnet,
                                   const float* __restrict__ bnet,
                                   const float* __restrict__ W1,
                                   const float* __restrict__ b1,
                                   const float* __restrict__ W2,
                                   const float* __restrict__ b2,
                                   const unsigned char* __restrict__ mask,
                                   float* __restrict__ out)
{
    __shared__ int   s_i0[BPW], s_i1[BPW], s_nv[BPW];
    __shared__ float s_feat[16][DOUT];
    __shared__ float s_h[16][HID];
    __shared__ float s_pred[16][NOUT];

    const int lane = threadIdx.x;
    const int base = blockIdx.x * BPW;

    // ---------- Phase 1: mask scan (first two valid indices + valid count) ----------
    if (lane < BPW) {
        const unsigned long long* mrow =
            (const unsigned long long*)(mask + (size_t)(base + lane) * SEQ);
        int nv = 0, i0 = 0, i1 = 0, found = 0;
        #pragma unroll
        for (int w = 0; w < SEQ / 8; ++w) {
            unsigned long long m = mrow[w];
            nv += __popcll(m);                 // bool bytes are 0x00/0x01
            while (m && found < 2) {
                int bit = __ffsll((long long)m) - 1;
                int pos = w * 8 + (bit >> 3);
                if (found == 0) i0 = pos; else i1 = pos;
                ++found;
                m &= (m - 1);                  // clear that byte's single bit
            }
        }
        s_i0[lane] = i0; s_i1[lane] = i1; s_nv[lane] = nv;
    }
    __syncthreads();

    // ---------- Phase 2: WMMA f32 GEMM: feats(16x64) x W_net(64x128) ----------
    // A fragment (16x4 f32): lane L -> M = L&15 ; VGPR0 = K(kbase+khi), VGPR1 = K+1
    const int mrow = lane & 15;
    const int bb   = mrow >> 1;
    const int sel  = mrow & 1;
    const int srow = sel ? s_i1[bb] : s_i0[bb];
    const float* __restrict__ arow =
        x + ((size_t)(base + bb) * SEQ + srow) * DIN;
    const int khi  = (lane >> 4) << 1;   // lanes 0-15 -> K+0/1, lanes 16-31 -> K+2/3
    const int ncol = lane & 15;          // B/C column within tile

    v8f acc[8] = {};
    #pragma unroll
    for (int c = 0; c < DIN / 4; ++c) {
        v2f a;
        a.x = arow[c * 4 + khi + 0];
        a.y = arow[c * 4 + khi + 1];
        #pragma unroll
        for (int t = 0; t < DOUT / 16; ++t) {
            const float* __restrict__ wp = Wnet + (c * 4 + khi) * DOUT + t * 16 + ncol;
            v2f b;
            b.x = wp[0];
            b.y = wp[DOUT];
            acc[t] = __builtin_amdgcn_wmma_f32_16x16x4_f32(
                false, a, false, b, (short)0, acc[t], false, false);
        }
    }

    // ---------- Phase 3: bias + spill C fragments to LDS ----------
    // C layout: VGPR r -> M = r (lanes 0-15) / M = r+8 (lanes 16-31), N = lane&15
    const int rowoff = (lane >> 4) * 8;
    #pragma unroll
    for (int t = 0; t < DOUT / 16; ++t) {
        const float bn = bnet[t * 16 + ncol];
        #pragma unroll
        for (int r = 0; r < 8; ++r)
            s_feat[rowoff + r][t * 16 + ncol] = acc[t][r] + bn;
    }
    __syncthreads();

    // ---------- Phase 4a: h = relu(feats @ W1 + b1) ----------
    {
        const int row = lane & 15;
        const int jg  = (lane >> 4) * (HID / 2);   // 16 hidden units per lane
        for (int j = 0; j < HID / 2; ++j) {
            float s = b1[jg + j];
            #pragma unroll 8
            for (int n = 0; n < DOUT; ++n)
                s += s_feat[row][n] * W1[n * HID + jg + j];
            s_h[row][jg + j] = fmaxf(s, 0.f);
        }
    }
    __syncthreads();

    // ---------- Phase 4b: preds = h @ W2 + b2 ----------
    if (lane < 16) {
        const int row = lane;
        #pragma unroll
        for (int o = 0; o < NOUT; ++o) {
            float s = b2[o];
            #pragma unroll
            for (int n = 0; n < HID; ++n)
                s += s_h[row][n] * W2[n * NOUT + o];
            s_pred[row][o] = s;
        }
    }
    __syncthreads();

    // ---------- Phase 5: n_valid gating + store (B,2,4) ----------
    if (lane < BPW) {
        const int nv = s_nv[lane];
        float* __restrict__ op = out + (size_t)(base + lane) * 2 * NOUT;
        #pragma unroll
        for (int o = 0; o < NOUT; ++o) {
            const float p0 = s_pred[2 * lane + 0][o];
            const float p1 = s_pred[2 * lane + 1][o];
            const float top1 = (nv >= 1) ? p0 : 0.f;
            const float top2 = (nv >= 2) ? p1 : ((nv == 1) ? p0 : 0.f);
            op[o]        = top1;
            op[NOUT + o] = top2;
        }
    }
}

extern "C" void kernel_launch(void* const* d_in, const int* in_sizes, int n_in,
                              void* d_out, int out_size, void* d_ws, size_t ws_size,
                              hipStream_t stream) {
    (void)in_sizes; (void)n_in; (void)out_size; (void)d_ws; (void)ws_size;
    const float* x    = (const float*)d_in[0];
    const float* Wnet = (const float*)d_in[1];
    const float* bnet = (const float*)d_in[2];
    const float* W1   = (const float*)d_in[3];
    const float* b1   = (const float*)d_in[4];
    const float* W2   = (const float*)d_in[5];
    const float* b2   = (const float*)d_in[6];
    const unsigned char* mask = (const unsigned char*)d_in[7];
    float* out = (float*)d_out;

    dim3 grid(B_TOT / BPW);   // 512 single-wave workgroups
    dim3 block(32);
    hipLaunchKernelGGL(fused_topk_transformer_kernel, grid, block, 0, stream,
                       x, Wnet, bnet, W1, b1, W2, b2, mask, out);
}